// DuvenaudGCNClassifier_35107062678356
// MI455X (gfx1250) — compile-verified
//
#include <hip/hip_runtime.h>
#include <hip/hip_bf16.h>

typedef __attribute__((ext_vector_type(2))) float v2f;
typedef __attribute__((ext_vector_type(8))) float v8f;

#define D_H 64

// epilogue modes for the fused GEMM
#define EPI_NONE 0
#define EPI_RELU 1
#define EPI_SOFTMAX 2

// ---------------------------------------------------------------- utilities
__global__ void zero_kernel(float* __restrict__ p, long long n) {
    long long i = (long long)blockIdx.x * blockDim.x + threadIdx.x;
    if (i < n) p[i] = 0.0f;
}

__global__ void degree_kernel(const int* __restrict__ dst,
                              float* __restrict__ deg, int nedges) {
    int e = blockIdx.x * blockDim.x + threadIdx.x;
    if (e < nedges) atomicAdd(&deg[dst[e]], 1.0f);
}

// 16 threads per edge, 4 floats each: coalesced float4 gather + f32 atomics.
__global__ void scatter_kernel(const float* __restrict__ feat,
                               const int* __restrict__ src,
                               const int* __restrict__ dst,
                               float* __restrict__ sum, int nedges) {
    long long t = (long long)blockIdx.x * blockDim.x + threadIdx.x;
    int e = (int)(t >> 4);
    int q = (int)(t & 15);
    if (e >= nedges) return;
    int s = src[e];
    int d = dst[e];
    float4 v = ((const float4*)(feat + (size_t)s * D_H))[q];
    float* o = sum + (size_t)d * D_H + q * 4;
    atomicAdd(o + 0, v.x);
    atomicAdd(o + 1, v.y);
    atomicAdd(o + 2, v.z);
    atomicAdd(o + 3, v.w);
}

// ---------------------- fused aggregate + WMMA GEMM + (relu | row-softmax)
// out[row, :] = epi( A[row, :] @ W(64x64) + bias ),
// A[row, :] = deg ? (deg[row]>0 ? Asum[row]/deg[row] : Afb[row]) : Asum[row]
// One wave per 16-row tile; C = 4 x (16x16) tiles via V_WMMA_F32_16X16X4_F32.
// Softmax epilogue: rows are partitioned by wave halves in the C/D layout
// (lanes 0-15: M=0..7, lanes 16-31: M=8..15; the 16 columns of each N-tile
// stripe across the 16 lanes of a half), so a full row reduces with
// __shfl_xor masks {8,4,2,1} which never cross the halves (wave32).
__global__ __launch_bounds__(256) void gemm64_wmma_kernel(
    const float* __restrict__ Asum, const float* __restrict__ Afb,
    const float* __restrict__ deg, const float* __restrict__ W,
    const float* __restrict__ bias, float* __restrict__ out,
    int nrows, int mode) {
    int wave = threadIdx.x >> 5;
    int lane = threadIdx.x & 31;
    int row0 = (blockIdx.x * 8 + wave) * 16;
    if (row0 >= nrows) return;  // wave-uniform

    int m     = lane & 15;
    int hi    = lane >> 4;        // 0: lanes 0-15, 1: lanes 16-31
    int khalf = hi << 1;          // K offset 0 or 2 within each 4-wide K step
    int row   = row0 + m;
    if (row >= nrows) row = nrows - 1;  // keep EXEC all-ones; stores guarded

    const float* abase = Asum;
    float ascale = 1.0f;
    if (deg) {
        float dv = deg[row];
        if (dv > 0.0f) ascale = 1.0f / dv;
        else           abase  = Afb;
    }

    // Preload the 16 A fragments (ISA 16x4 f32 A layout: lane<16 holds K=0,1;
    // lane>=16 holds K=2,3 of each 4-wide K step, row M = lane&15).
    const float* arow = abase + (size_t)row * D_H + khalf;
    v2f afrag[16];
#pragma unroll
    for (int i = 0; i < 16; ++i) {
        afrag[i].x = arow[i * 4 + 0] * ascale;
        afrag[i].y = arow[i * 4 + 1] * ascale;
    }

    v8f accs[4];
#pragma unroll
    for (int nt = 0; nt < 4; ++nt) {
        int n = nt * 16 + (lane & 15);
        float bv = bias[n];
        v8f acc = {bv, bv, bv, bv, bv, bv, bv, bv};
#pragma unroll
        for (int i = 0; i < 16; ++i) {
            int k = i * 4 + khalf;
            v2f bfrag;
            bfrag.x = W[(k + 0) * D_H + n];   // B 4x16: row K striped over lanes
            bfrag.y = W[(k + 1) * D_H + n];
            acc = __builtin_amdgcn_wmma_f32_16x16x4_f32(
                false, afrag[i], false, bfrag, (short)0, acc, false, false);
        }
        accs[nt] = acc;
    }

    if (mode == EPI_RELU) {
#pragma unroll
        for (int nt = 0; nt < 4; ++nt)
#pragma unroll
            for (int r = 0; r < 8; ++r) accs[nt][r] = fmaxf(accs[nt][r], 0.0f);
    } else if (mode == EPI_SOFTMAX) {
        float mx[8];
#pragma unroll
        for (int r = 0; r < 8; ++r)
            mx[r] = fmaxf(fmaxf(accs[0][r], accs[1][r]),
                          fmaxf(accs[2][r], accs[3][r]));
#pragma unroll
        for (int mask = 8; mask >= 1; mask >>= 1)
#pragma unroll
            for (int r = 0; r < 8; ++r)
                mx[r] = fmaxf(mx[r], __shfl_xor(mx[r], mask, 32));
        float sum[8] = {0.f, 0.f, 0.f, 0.f, 0.f, 0.f, 0.f, 0.f};
#pragma unroll
        for (int nt = 0; nt < 4; ++nt)
#pragma unroll
            for (int r = 0; r < 8; ++r) {
                float e = __expf(accs[nt][r] - mx[r]);
                accs[nt][r] = e;
                sum[r] += e;
            }
#pragma unroll
        for (int mask = 8; mask >= 1; mask >>= 1)
#pragma unroll
            for (int r = 0; r < 8; ++r) sum[r] += __shfl_xor(sum[r], mask, 32);
#pragma unroll
        for (int r = 0; r < 8; ++r) {
            float inv = 1.0f / sum[r];
#pragma unroll
            for (int nt = 0; nt < 4; ++nt) accs[nt][r] *= inv;
        }
    }

    int mbase = row0 + (hi << 3);  // C/D: lanes 16-31 hold rows M=8..15
#pragma unroll
    for (int nt = 0; nt < 4; ++nt) {
        int n = nt * 16 + (lane & 15);
#pragma unroll
        for (int r = 0; r < 8; ++r) {
            int orow = mbase + r;
            if (orow < nrows) out[(size_t)orow * D_H + n] = accs[nt][r];
        }
    }
}

// ------------------------------------------------------------- graph pooling
// fg[g]  += f[n] ; nfg[g] += f[n] + h2s[n]   (new_f = f + h2s)
__global__ void pool_kernel(const float* __restrict__ f,
                            const float* __restrict__ h2,
                            const int* __restrict__ gid,
                            float* __restrict__ fg, float* __restrict__ nfg,
                            int nnodes) {
    long long t = (long long)blockIdx.x * blockDim.x + threadIdx.x;
    int n = (int)(t >> 4);
    int q = (int)(t & 15);
    if (n >= nnodes) return;
    int g = gid[n];
    float4 fv = ((const float4*)(f  + (size_t)n * D_H))[q];
    float4 hv = ((const float4*)(h2 + (size_t)n * D_H))[q];
    float* pf = fg  + (size_t)g * D_H + q * 4;
    float* pn = nfg + (size_t)g * D_H + q * 4;
    atomicAdd(pf + 0, fv.x); atomicAdd(pf + 1, fv.y);
    atomicAdd(pf + 2, fv.z); atomicAdd(pf + 3, fv.w);
    atomicAdd(pn + 0, fv.x + hv.x); atomicAdd(pn + 1, fv.y + hv.y);
    atomicAdd(pn + 2, fv.z + hv.z); atomicAdd(pn + 3, fv.w + hv.w);
}

// --------------------------------------------------------------- final head
// x = relu(concat(fg, nfg) @ Wd + bd) ; out = x @ Wc + bc   (per graph block)
__global__ __launch_bounds__(64) void head_kernel(
    const float* __restrict__ fg, const float* __restrict__ nfg,
    const float* __restrict__ Wd, const float* __restrict__ bd,
    const float* __restrict__ Wc, const float* __restrict__ bc,
    float* __restrict__ out) {
    __shared__ float xs[D_H];
    int b = blockIdx.x;
    int d = threadIdx.x;
    float acc = bd[d];
    const float* fr = fg  + (size_t)b * D_H;
    const float* nr = nfg + (size_t)b * D_H;
#pragma unroll 4
    for (int k = 0; k < D_H; ++k) acc += fr[k] * Wd[k * D_H + d];
#pragma unroll 4
    for (int k = 0; k < D_H; ++k) acc += nr[k] * Wd[(D_H + k) * D_H + d];
    xs[d] = fmaxf(acc, 0.0f);
    __syncthreads();
    if (d == 0) {
        float o = bc[0];
        for (int k = 0; k < D_H; ++k) o += xs[k] * Wc[k];
        out[b] = o;
    }
}

// ----------------------------------------------------------------- launcher
extern "C" void kernel_launch(void* const* d_in, const int* in_sizes, int n_in,
                              void* d_out, int out_size, void* d_ws,
                              size_t ws_size, hipStream_t stream) {
    const float* feat = (const float*)d_in[0];
    const int*   src  = (const int*)d_in[1];
    const int*   dst  = (const int*)d_in[2];
    const int*   gid  = (const int*)d_in[3];
    const float* W1  = (const float*)d_in[4];
    const float* b1  = (const float*)d_in[5];
    const float* Ws1 = (const float*)d_in[6];
    const float* bs1 = (const float*)d_in[7];
    const float* W2  = (const float*)d_in[8];
    const float* b2  = (const float*)d_in[9];
    const float* Ws2 = (const float*)d_in[10];
    const float* bs2 = (const float*)d_in[11];
    const float* Wd  = (const float*)d_in[12];
    const float* bd  = (const float*)d_in[13];
    const float* Wc  = (const float*)d_in[14];
    const float* bc  = (const float*)d_in[15];
    float* outp = (float*)d_out;

    const int nnodes  = in_sizes[0] / D_H;   // 100000
    const int nedges  = in_sizes[1];         // 1600000
    const int ngraphs = out_size;            // 128

    // workspace carve-out
    char* ws = (char*)d_ws;
    size_t off = 0;
    auto carve = [&](size_t bytes) {
        size_t cur = off;
        off += (bytes + 255) & ~(size_t)255;
        return (float*)(ws + cur);
    };
    const size_t nodeb = (size_t)nnodes * D_H * sizeof(float);
    float* deg    = carve((size_t)nnodes * sizeof(float));
    float* sumbuf = carve(nodeb);
    float* t1     = carve(nodeb);
    float* t2     = carve(nodeb);   // holds f after softmax1
    float* fg     = carve((size_t)ngraphs * D_H * sizeof(float));
    float* nfg    = carve((size_t)ngraphs * D_H * sizeof(float));
    (void)ws_size; (void)n_in;

    const int TB = 256;
    const long long nodeElems = (long long)nnodes * D_H;
    dim3 zg1((unsigned)((nnodes + TB - 1) / TB));
    dim3 zgN((unsigned)((nodeElems + TB - 1) / TB));
    dim3 edgeg((unsigned)(((long long)nedges * 16 + TB - 1) / TB));
    dim3 degg((unsigned)((nedges + TB - 1) / TB));
    int tiles = (nnodes + 15) / 16;
    dim3 gemmg((unsigned)((tiles + 7) / 8));
    dim3 poolg((unsigned)(((long long)nnodes * 16 + TB - 1) / TB));
    dim3 zgg((unsigned)((ngraphs * D_H + TB - 1) / TB));

    // -------- layer 1: mean-aggregate(feat) -> W1/relu -> Ws1+softmax
    zero_kernel<<<zg1, TB, 0, stream>>>(deg, nnodes);
    zero_kernel<<<zgN, TB, 0, stream>>>(sumbuf, nodeElems);
    degree_kernel<<<degg, TB, 0, stream>>>(dst, deg, nedges);
    scatter_kernel<<<edgeg, TB, 0, stream>>>(feat, src, dst, sumbuf, nedges);
    gemm64_wmma_kernel<<<gemmg, TB, 0, stream>>>(sumbuf, feat, deg, W1, b1, t1,
                                                 nnodes, EPI_RELU);
    gemm64_wmma_kernel<<<gemmg, TB, 0, stream>>>(t1, t1, nullptr, Ws1, bs1, t2,
                                                 nnodes, EPI_SOFTMAX);  // t2 = f

    // -------- layer 2: mean-aggregate(f) -> W2/relu -> Ws2+softmax
    zero_kernel<<<zgN, TB, 0, stream>>>(sumbuf, nodeElems);
    scatter_kernel<<<edgeg, TB, 0, stream>>>(t2, src, dst, sumbuf, nedges);
    gemm64_wmma_kernel<<<gemmg, TB, 0, stream>>>(sumbuf, t2, deg, W2, b2, t1,
                                                 nnodes, EPI_RELU);
    gemm64_wmma_kernel<<<gemmg, TB, 0, stream>>>(t1, t1, nullptr, Ws2, bs2,
                                                 sumbuf, nnodes, EPI_SOFTMAX);
    // sumbuf = h2s

    // -------- pooling + head
    zero_kernel<<<zgg, TB, 0, stream>>>(fg, ngraphs * D_H);
    zero_kernel<<<zgg, TB, 0, stream>>>(nfg, ngraphs * D_H);
    pool_kernel<<<poolg, TB, 0, stream>>>(t2, sumbuf, gid, fg, nfg, nnodes);
    head_kernel<<<dim3((unsigned)ngraphs), 64, 0, stream>>>(fg, nfg, Wd, bd, Wc,
                                                            bc, outp);
}